// FilterDetections2_79018808312550
// MI455X (gfx1250) — compile-verified
//
#include <hip/hip_runtime.h>
#include <stdint.h>

typedef unsigned int u32;
typedef unsigned long long u64;
typedef __attribute__((ext_vector_type(4))) int v4i;

#define GLOBAL_AS __attribute__((address_space(1)))
#define LDS_AS    __attribute__((address_space(3)))

#define BATCH   8
#define NANCH   100000
#define NCLS    80
#define NCTOT   (NANCH * NCLS)      /* 8,000,000 per batch */
#define KDET    300
#define NBINS   32768
#define CAP1    6144
#define THRESH  0.01f

#define HAVE_ASYNC_LDS __has_builtin(__builtin_amdgcn_global_load_async_to_lds_b128)

#if __has_builtin(__builtin_amdgcn_s_wait_asynccnt)
#define WAIT_ASYNC0() __builtin_amdgcn_s_wait_asynccnt(0)
#define WAIT_ASYNC1() __builtin_amdgcn_s_wait_asynccnt(1)
#else
#define WAIT_ASYNC0() asm volatile("s_wait_asynccnt 0x0" ::: "memory")
#define WAIT_ASYNC1() asm volatile("s_wait_asynccnt 0x1" ::: "memory")
#endif

__device__ __forceinline__ void hist_accum(u32* __restrict__ h, float s) {
    if (s > THRESH) atomicAdd(&h[(int)fminf(s * 32768.0f, 32767.0f)], 1u);
}

// ---------------------------------------------------------------------------
// Pass 1: value-linear histogram of scores > THRESH, per batch.
// bin = trunc(min(score * 32768, 32767)) is monotone in score, so the bin of
// the 300th-largest value is an exact coarse cutoff. Value-linear bins spread
// uniform data evenly -> global atomics are contention-free.
//
// CDNA5 path: the 256 MB stream is staged through LDS with ASYNCcnt-tracked
// global_load_async_to_lds_b128, double-buffered per wave (each lane copies
// 16 B into its own LDS slot; all bounds are wave-uniform since every stride
// is a multiple of 32, so no barriers are needed).
// ---------------------------------------------------------------------------
#if HAVE_ASYNC_LDS
__global__ __launch_bounds__(256)
void hist_kernel(const float* __restrict__ cls, u32* __restrict__ hist) {
    __shared__ __align__(16) float4 stage[2][256];   // 8 KB, per-thread slots
    const int total4 = BATCH * NCTOT / 4;            // 16,000,000 float4s
    const int stride = gridDim.x * blockDim.x;       // multiple of 32
    const int tid = threadIdx.x;

    GLOBAL_AS v4i* gbase = (GLOBAL_AS v4i*)cls;
    LDS_AS v4i* l0 = (LDS_AS v4i*)&stage[0][tid];
    LDS_AS v4i* l1 = (LDS_AS v4i*)&stage[1][tid];

    int v = blockIdx.x * blockDim.x + tid;
    int buf = 0;
    if (v < total4)
        __builtin_amdgcn_global_load_async_to_lds_b128(gbase + v, l0, 0, 0);

    for (; v < total4; v += stride) {
        const int vn = v + stride;
        if (vn < total4) {
            __builtin_amdgcn_global_load_async_to_lds_b128(gbase + vn, buf ? l0 : l1, 0, 0);
            WAIT_ASYNC1();                           // current buffer complete
        } else {
            WAIT_ASYNC0();
        }
        asm volatile("" ::: "memory");
        const float4 f = stage[buf][tid];
        const int b = (v * 4) / NCTOT;
        u32* __restrict__ h = hist + b * NBINS;
        hist_accum(h, f.x); hist_accum(h, f.y);
        hist_accum(h, f.z); hist_accum(h, f.w);
        buf ^= 1;
    }
}
#else
__global__ __launch_bounds__(256)
void hist_kernel(const float* __restrict__ cls, u32* __restrict__ hist) {
    const int total4 = BATCH * NCTOT / 4;
    const int stride = gridDim.x * blockDim.x;
    for (int v = blockIdx.x * blockDim.x + threadIdx.x; v < total4; v += stride) {
        const float4 f = ((const float4*)cls)[v];
        const int b = (v * 4) / NCTOT;
        u32* __restrict__ h = hist + b * NBINS;
        hist_accum(h, f.x); hist_accum(h, f.y);
        hist_accum(h, f.z); hist_accum(h, f.w);
    }
}
#endif

// ---------------------------------------------------------------------------
// Pass 2: per batch, scan bins from the top; find bin b1 where the cumulative
// count first reaches KDET. If total survivors < KDET, b1 = 0 (keep all).
// ---------------------------------------------------------------------------
__global__ __launch_bounds__(1024)
void scan_kernel(const u32* __restrict__ hist, u32* __restrict__ cut) {
    __shared__ u32 seg[1024];
    __shared__ int s_ts;
    __shared__ u32 s_cum;
    const int b = blockIdx.x;
    const u32* __restrict__ h = hist + b * NBINS;
    const int t = threadIdx.x;

    // segment t covers bins [NBINS-1-32t .. NBINS-32-32t], descending
    const int hi = NBINS - 1 - t * 32;
    u32 sum = 0;
    for (int i = 0; i < 32; ++i) sum += h[hi - i];
    seg[t] = sum;
    __syncthreads();

    if (t == 0) {
        u32 cum = 0;
        int ts = -1;
        for (int s = 0; s < 1024; ++s) {
            if (cum + seg[s] >= (u32)KDET) { ts = s; break; }
            cum += seg[s];
        }
        s_ts = ts;
        s_cum = cum;
    }
    __syncthreads();

    const int ts = s_ts;
    if (ts < 0) {
        if (t == 0) cut[b] = 0;          // fewer than KDET survivors: keep all
    } else if (t == ts) {
        u32 cum = s_cum;
        const int h2 = NBINS - 1 - ts * 32;
        u32 b1 = 0;
        for (int i = 0; i < 32; ++i) {
            const u32 c = h[h2 - i];
            if (cum + c >= (u32)KDET) { b1 = (u32)(h2 - i); break; }
            cum += c;
        }
        cut[b] = b1;
    }
}

// ---------------------------------------------------------------------------
// Pass 3: compact candidates (score > THRESH and bin >= b1) into 64-bit keys.
// flat index in class-major order: j = c*N + n  (memory is [n][c] row-major).
// key = bits(score) << 23 | (0x7FFFFF - j)  => larger key == (value desc, j asc)
// Same ASYNCcnt double-buffered staging as the histogram pass.
// ---------------------------------------------------------------------------
__device__ __forceinline__ void compact_elem(float s, int b, int cl, int n, u32 b1,
                                             u64* __restrict__ cand,
                                             u32* __restrict__ cnt) {
    if (s > THRESH) {
        const u32 bin = (u32)(int)fminf(s * 32768.0f, 32767.0f);
        if (bin >= b1) {
            const u32 bits = __float_as_uint(s);
            const u32 flat = (u32)cl * (u32)NANCH + (u32)n;
            const u64 key  = ((u64)bits << 23) | (u64)(8388607u - flat);
            const u32 pos  = atomicAdd(&cnt[b], 1u);
            if (pos < (u32)CAP1) cand[(size_t)b * CAP1 + pos] = key;
        }
    }
}

#if HAVE_ASYNC_LDS
__global__ __launch_bounds__(256)
void compact_kernel(const float* __restrict__ cls, const u32* __restrict__ cut,
                    u64* __restrict__ cand, u32* __restrict__ cnt) {
    __shared__ __align__(16) float4 stage[2][256];
    const int total4 = BATCH * NCTOT / 4;
    const int stride = gridDim.x * blockDim.x;       // multiple of 32
    const int tid = threadIdx.x;

    GLOBAL_AS v4i* gbase = (GLOBAL_AS v4i*)cls;
    LDS_AS v4i* l0 = (LDS_AS v4i*)&stage[0][tid];
    LDS_AS v4i* l1 = (LDS_AS v4i*)&stage[1][tid];

    int v = blockIdx.x * blockDim.x + tid;
    int buf = 0;
    if (v < total4)
        __builtin_amdgcn_global_load_async_to_lds_b128(gbase + v, l0, 0, 0);

    for (; v < total4; v += stride) {
        const int vn = v + stride;
        if (vn < total4) {
            __builtin_amdgcn_global_load_async_to_lds_b128(gbase + vn, buf ? l0 : l1, 0, 0);
            WAIT_ASYNC1();
        } else {
            WAIT_ASYNC0();
        }
        asm volatile("" ::: "memory");
        const float4 f = stage[buf][tid];
        const int p   = v * 4;
        const int b   = p / NCTOT;
        const int rem = p - b * NCTOT;
        const int n   = rem / NCLS;
        const int c   = rem - n * NCLS;      // multiple of 4; never crosses a row
        const u32 b1  = cut[b];
        compact_elem(f.x, b, c + 0, n, b1, cand, cnt);
        compact_elem(f.y, b, c + 1, n, b1, cand, cnt);
        compact_elem(f.z, b, c + 2, n, b1, cand, cnt);
        compact_elem(f.w, b, c + 3, n, b1, cand, cnt);
        buf ^= 1;
    }
}
#else
__global__ __launch_bounds__(256)
void compact_kernel(const float* __restrict__ cls, const u32* __restrict__ cut,
                    u64* __restrict__ cand, u32* __restrict__ cnt) {
    const int total4 = BATCH * NCTOT / 4;
    const int stride = gridDim.x * blockDim.x;
    for (int v = blockIdx.x * blockDim.x + threadIdx.x; v < total4; v += stride) {
        if (v + stride < total4)
            __builtin_prefetch(&((const float4*)cls)[v + stride], 0, 0);
        const float4 f = ((const float4*)cls)[v];
        const int p   = v * 4;
        const int b   = p / NCTOT;
        const int rem = p - b * NCTOT;
        const int n   = rem / NCLS;
        const int c   = rem - n * NCLS;
        const u32 b1  = cut[b];
        compact_elem(f.x, b, c + 0, n, b1, cand, cnt);
        compact_elem(f.y, b, c + 1, n, b1, cand, cnt);
        compact_elem(f.z, b, c + 2, n, b1, cand, cnt);
        compact_elem(f.w, b, c + 3, n, b1, cand, cnt);
    }
}
#endif

// ---------------------------------------------------------------------------
// Pass 4: one block per batch. Exact rank of each unique key = #keys greater.
// Keys with rank < KDET scatter straight into sorted output slots; everything
// else stays at the -1 fill. Deterministic regardless of append order.
// ---------------------------------------------------------------------------
__global__ __launch_bounds__(1024)
void final_kernel(const float* __restrict__ boxes, const u64* __restrict__ cand,
                  const u32* __restrict__ cnt, float* __restrict__ out_boxes,
                  float* __restrict__ out_scores, int* __restrict__ out_labels) {
    __shared__ u64 keys[CAP1];                  // 48 KB
    const int b = blockIdx.x;
    const int t = threadIdx.x;
    u32 M = cnt[b];
    if (M > (u32)CAP1) M = CAP1;

    for (u32 i = t; i < M; i += blockDim.x) keys[i] = cand[(size_t)b * CAP1 + i];

    // -1 fill of this batch's outputs
    for (int i = t; i < KDET; i += blockDim.x) {
        out_scores[b * KDET + i] = -1.0f;
        out_labels[b * KDET + i] = -1;
        ((float4*)out_boxes)[b * KDET + i] = make_float4(-1.0f, -1.0f, -1.0f, -1.0f);
    }
    __threadfence();
    __syncthreads();

    for (u32 i = t; i < M; i += blockDim.x) {
        const u64 k = keys[i];
        u32 r = 0;
        for (u32 j = 0; j < M; ++j) r += (keys[j] > k) ? 1u : 0u;
        if (r < (u32)KDET) {
            const u32 bits   = (u32)(k >> 23);
            const u32 flat   = 8388607u - (u32)(k & 0x7FFFFFu);
            const u32 anchor = flat % (u32)NANCH;
            const u32 label  = flat / (u32)NANCH;
            out_scores[b * KDET + r] = __uint_as_float(bits);
            out_labels[b * KDET + r] = (int)label;
            ((float4*)out_boxes)[b * KDET + r] =
                ((const float4*)boxes)[(size_t)b * NANCH + anchor];
        }
    }
}

// ---------------------------------------------------------------------------
extern "C" void kernel_launch(void* const* d_in, const int* in_sizes, int n_in,
                              void* d_out, int out_size, void* d_ws, size_t ws_size,
                              hipStream_t stream) {
    (void)in_sizes; (void)n_in; (void)out_size; (void)ws_size;

    const float* boxes = (const float*)d_in[0];   // [B, N, 4]
    const float* cls   = (const float*)d_in[1];   // [B, N, C]

    char* ws = (char*)d_ws;
    u32* hist = (u32*)ws;                                     // B*NBINS u32 = 1 MB
    u32* cnt1 = (u32*)(ws + (size_t)BATCH * NBINS * 4);       // B u32
    u32* cut1 = cnt1 + BATCH;                                 // B u32
    u64* cand = (u64*)(ws + (size_t)BATCH * NBINS * 4 + 2u * BATCH * 4);  // 8-aligned

    // zero hist + counters every call (graph replays include this)
    (void)hipMemsetAsync(ws, 0, (size_t)BATCH * NBINS * 4 + 2u * BATCH * 4, stream);

    hist_kernel   <<<3072, 256, 0, stream>>>(cls, hist);
    scan_kernel   <<<BATCH, 1024, 0, stream>>>(hist, cut1);
    compact_kernel<<<3072, 256, 0, stream>>>(cls, cut1, cand, cnt1);

    float* out_boxes  = (float*)d_out;                       // B*KDET*4
    float* out_scores = out_boxes + BATCH * KDET * 4;        // B*KDET
    int*   out_labels = (int*)(out_scores + BATCH * KDET);   // B*KDET (int32)

    final_kernel<<<BATCH, 1024, 0, stream>>>(boxes, cand, cnt1,
                                             out_boxes, out_scores, out_labels);
}